// FISLayerFixedTree_3719441679079
// MI455X (gfx1250) — compile-verified
//
#include <hip/hip_runtime.h>
#include <hip/hip_bf16.h>

typedef __attribute__((ext_vector_type(2))) float v2f;
typedef __attribute__((ext_vector_type(8))) float v8f;

#define CIN   64
#define COUT  128
#define HW    128          // H == W == 128
#define XSTR  66           // LDS stride for transposed X tile [n][i] (even -> 8B aligned rows)
#define LSTR  133          // LDS stride for leaf/mid tile [o][n] (odd -> conflict-free scan)

// One 16x16 fp32 output tile of (alpha[128x64] * X[64x128]) via 16 chained
// V_WMMA_F32_16X16X4_F32. sXT holds X transposed: sXT[n*XSTR + i] = X[i][n].
// Per ISA layouts: A frag lane m=lane%16, K pair selected by lane/16;
// B frag identical formula on the transposed X; C/D row = r + 8*(lane/16).
__device__ __forceinline__ v8f gemm_tile(const float* __restrict__ alpha, int oBase,
                                         const float* sXT, int nCol,
                                         int l16, int half) {
    v8f acc = {};
#pragma unroll
    for (int k = 0; k < CIN; k += 4) {
        v2f af = *(const v2f*)&alpha[(oBase + l16) * CIN + k + 2 * half];
        v2f bf = *(const v2f*)&sXT[nCol * XSTR + k + 2 * half];
        acc = __builtin_amdgcn_wmma_f32_16x16x4_f32(
            /*neg_a=*/false, af, /*neg_b=*/false, bf,
            /*c_mod=*/(short)0, acc, /*reuse_a=*/false, /*reuse_b=*/false);
    }
    return acc;
}

// Kernel 1: per (b,h) row -> leaf = a3*X, strict discounted scan along W,
// mid = (a2*X) * c_w, stored into d_out (used as the mid buffer).
__global__ __launch_bounds__(256, 1)
void fis_k1(const float* __restrict__ x, const float* __restrict__ alpha2,
            const float* __restrict__ alpha3, const float* __restrict__ disc,
            float* __restrict__ mid) {
    __shared__ __align__(16) float sXT[HW * XSTR];   // [w][i]
    __shared__ __align__(16) float sL [COUT * LSTR]; // [o][w]

    const int b = blockIdx.x >> 7;
    const int h = blockIdx.x & 127;
    const int tid  = threadIdx.x;
    const int lane = tid & 31;
    const int wv   = tid >> 5;       // 8 waves
    const int half = lane >> 4;
    const int l16  = lane & 15;
    const int oBase = wv * 16;

    // Stage x[b, :, h, :] into LDS, transposed to [w][i].
    const float* xb = x + (((size_t)b * CIN) * HW + h) * HW;
    for (int idx = tid; idx < CIN * HW; idx += 256) {
        int i = idx >> 7, w = idx & 127;
        sXT[w * XSTR + i] = xb[(size_t)i * (HW * HW) + w];
    }
    __syncthreads();

    // leaf = alpha3 * X  -> sL
    for (int wt = 0; wt < 8; ++wt) {
        int n = wt * 16 + l16;
        v8f acc = gemm_tile(alpha3, oBase, sXT, n, l16, half);
#pragma unroll
        for (int r = 0; r < 8; ++r)
            sL[(oBase + r + 8 * half) * LSTR + n] = acc[r];
    }
    __syncthreads();

    // Strict discounted cumsum along W, in place: row[w] <- s_{w-1}, s_{-1}=0.
    if (tid < COUT) {
        const float d = disc[tid];
        float s = 0.0f;
        float* row = &sL[tid * LSTR];
#pragma unroll 4
        for (int w = 0; w < HW; ++w) {
            float v = row[w];
            row[w]  = s;
            s = fmaf(d, s, v);
        }
    }
    __syncthreads();

    // mid = (alpha2 * X) * c_w  -> global (d_out reused as mid buffer)
    float* mb = mid + (((size_t)b * COUT) * HW + h) * HW;
    for (int wt = 0; wt < 8; ++wt) {
        int n = wt * 16 + l16;
        v8f acc = gemm_tile(alpha2, oBase, sXT, n, l16, half);
#pragma unroll
        for (int r = 0; r < 8; ++r) {
            int o = oBase + r + 8 * half;
            mb[(size_t)o * (HW * HW) + n] = acc[r] * sL[o * LSTR + n];
        }
    }
}

// Kernel 2: per (b,w) column -> strict discounted scan of mid along H,
// out = (a1*X) * c_h, written in place over the mid buffer (d_out).
__global__ __launch_bounds__(256, 1)
void fis_k2(const float* __restrict__ x, const float* __restrict__ alpha1,
            const float* __restrict__ disc, float* __restrict__ io) {
    __shared__ __align__(16) float sXT[HW * XSTR];   // [h][i]
    __shared__ __align__(16) float sM [COUT * LSTR]; // [o][h]

    const int b = blockIdx.x >> 7;
    const int w = blockIdx.x & 127;
    const int tid  = threadIdx.x;
    const int lane = tid & 31;
    const int wv   = tid >> 5;
    const int half = lane >> 4;
    const int l16  = lane & 15;
    const int oBase = wv * 16;

    // Stage x[b, :, :, w] into LDS transposed to [h][i].
    const float* xb = x + ((size_t)b * CIN) * (HW * HW) + w;
    for (int idx = tid; idx < CIN * HW; idx += 256) {
        int i = idx >> 7, h = idx & 127;
        sXT[h * XSTR + i] = xb[(size_t)i * (HW * HW) + (size_t)h * HW];
    }
    // Stage mid[b, :, :, w] into LDS as [o][h].
    float* iob = io + ((size_t)b * COUT) * (HW * HW) + w;
    for (int idx = tid; idx < COUT * HW; idx += 256) {
        int o = idx >> 7, h = idx & 127;
        sM[o * LSTR + h] = iob[(size_t)o * (HW * HW) + (size_t)h * HW];
    }
    __syncthreads();

    // Strict discounted cumsum along H, in place.
    if (tid < COUT) {
        const float d = disc[tid];
        float s = 0.0f;
        float* row = &sM[tid * LSTR];
#pragma unroll 4
        for (int h = 0; h < HW; ++h) {
            float v = row[h];
            row[h]  = s;
            s = fmaf(d, s, v);
        }
    }
    __syncthreads();

    // out = (alpha1 * X) * c_h -> same addresses we read mid from.
    for (int ht = 0; ht < 8; ++ht) {
        int n = ht * 16 + l16;
        v8f acc = gemm_tile(alpha1, oBase, sXT, n, l16, half);
#pragma unroll
        for (int r = 0; r < 8; ++r) {
            int o = oBase + r + 8 * half;
            iob[(size_t)o * (HW * HW) + (size_t)n * HW] = acc[r] * sM[o * LSTR + n];
        }
    }
}

extern "C" void kernel_launch(void* const* d_in, const int* in_sizes, int n_in,
                              void* d_out, int out_size, void* d_ws, size_t ws_size,
                              hipStream_t stream) {
    const float* x      = (const float*)d_in[0];
    const float* alpha1 = (const float*)d_in[1];
    const float* alpha2 = (const float*)d_in[2];
    const float* alpha3 = (const float*)d_in[3];
    const float* disc   = (const float*)d_in[4];
    float* out = (float*)d_out;

    const int B = 16;
    dim3 block(256);
    dim3 grid(B * HW);  // 2048 blocks: one per (b,h) / (b,w)

    fis_k1<<<grid, block, 0, stream>>>(x, alpha2, alpha3, disc, out);
    fis_k2<<<grid, block, 0, stream>>>(x, alpha1, disc, out);
}